// EmbLoss_v1_64957085385421
// MI455X (gfx1250) — compile-verified
//
#include <hip/hip_runtime.h>
#include <math.h>

// ---------------------------------------------------------------------------
// EmbLoss (PAN-style discriminative embedding loss) for MI455X / gfx1250.
// Memory-bound segment reduction + one genuine WMMA stage:
//   Gram = M * M^T  (M = 32x4 f32 means) via V_WMMA_F32_16X16X4_F32.
// ---------------------------------------------------------------------------

#define HWPIX   (640 * 640)   // H*W per image
#define DFEAT   4
#define MAXL    33            // num_labels upper bound (ref: 33)
#define DELTA_V 0.5f
#define DELTA_D 1.5f

// per-batch workspace layout (floats)
#define OFF_SUMS 0            // 33*4
#define OFF_CNTS 132          // 33   (counts under inst_k)
#define OFF_CNTI 165          // 33   (counts under inst)
#define OFF_VALS 198          // 33   (segment sums of val)
#define OFF_MEAN 231          // 33*4
#define OFF_LDR  363          // 1    (l_dis + l_reg)
#define WS_PER_B 384

typedef __attribute__((ext_vector_type(2))) float v2f;
typedef __attribute__((ext_vector_type(8))) float v8f;

__global__ void k_zero(float* ws, int n) {
  for (int i = blockIdx.x * blockDim.x + threadIdx.x; i < n;
       i += gridDim.x * blockDim.x)
    ws[i] = 0.0f;
}

// Pass 1: per-label embedding sums + counts (LDS bins -> global atomics).
__global__ void k_segsum(const float* __restrict__ emb,
                         const int* __restrict__ inst,
                         const float* __restrict__ kern,
                         const float* __restrict__ tmask,
                         const int* __restrict__ nlp,
                         float* __restrict__ ws) {
  const int b = blockIdx.y;
  int NL = *nlp; NL = NL < 1 ? 1 : (NL > MAXL ? MAXL : NL);

  __shared__ float s_sums[MAXL * 4];
  __shared__ float s_cnts[MAXL];
  __shared__ float s_cnti[MAXL];
  for (int i = threadIdx.x; i < MAXL * 4; i += blockDim.x) s_sums[i] = 0.0f;
  for (int i = threadIdx.x; i < MAXL; i += blockDim.x) {
    s_cnts[i] = 0.0f;
    s_cnti[i] = 0.0f;
  }
  __syncthreads();

  const float* e  = emb  + (size_t)b * DFEAT * HWPIX;
  const int*   ib = inst + (size_t)b * HWPIX;
  const float* kb = kern + (size_t)b * HWPIX;
  const float* tb = tmask + (size_t)b * HWPIX;

  const int stride = gridDim.x * blockDim.x;
  for (int n = blockIdx.x * blockDim.x + threadIdx.x; n < HWPIX; n += stride) {
    int lab = ib[n];
    int i2  = (tb[n] > 0.5f) ? lab : 0;
    if (i2 < 0 || i2 >= NL) i2 = 0;
    int ik  = (kb[n] > 0.5f) ? i2 : 0;

    atomicAdd(&s_cnti[i2], 1.0f);
    atomicAdd(&s_cnts[ik], 1.0f);
    if (ik > 0) {
      atomicAdd(&s_sums[ik * 4 + 0], e[0 * HWPIX + n]);
      atomicAdd(&s_sums[ik * 4 + 1], e[1 * HWPIX + n]);
      atomicAdd(&s_sums[ik * 4 + 2], e[2 * HWPIX + n]);
      atomicAdd(&s_sums[ik * 4 + 3], e[3 * HWPIX + n]);
    }
  }
  __syncthreads();

  float* wsb = ws + (size_t)b * WS_PER_B;
  for (int i = threadIdx.x; i < NL * 4; i += blockDim.x)
    atomicAdd(&wsb[OFF_SUMS + i], s_sums[i]);
  for (int i = threadIdx.x; i < NL; i += blockDim.x) {
    atomicAdd(&wsb[OFF_CNTS + i], s_cnts[i]);
    atomicAdd(&wsb[OFF_CNTI + i], s_cnti[i]);
  }
}

// Stage 2: means + pairwise-distance loss via WMMA f32 16x16x4 Gram matrix.
// One wave (32 lanes) per batch image; EXEC all ones around the WMMAs.
__global__ void __launch_bounds__(32)
k_mean_pdist(const int* __restrict__ nlp, float* __restrict__ ws) {
  const int b    = blockIdx.x;
  const int lane = threadIdx.x;  // 0..31
  int NL = *nlp; NL = NL < 1 ? 1 : (NL > MAXL ? MAXL : NL);
  const int K = NL - 1;          // 32 for the reference

  float* wsb = ws + (size_t)b * WS_PER_B;

  __shared__ float s_m[32][4];   // means for labels 1..32 (zero-padded)
  __shared__ float s_n2[32];     // squared norms per row
  __shared__ float s_red[2];     // {sum vald, sum reg}

  if (lane < 2) s_red[lane] = 0.0f;

  // compute means (mean[0] forced to 0), publish to ws for pass 2
  for (int s = lane; s < MAXL; s += 32) {
    float m0 = 0.f, m1 = 0.f, m2 = 0.f, m3 = 0.f;
    if (s > 0 && s < NL) {
      float inv = 1.0f / fmaxf(wsb[OFF_CNTS + s], 1.0f);
      m0 = wsb[OFF_SUMS + s * 4 + 0] * inv;
      m1 = wsb[OFF_SUMS + s * 4 + 1] * inv;
      m2 = wsb[OFF_SUMS + s * 4 + 2] * inv;
      m3 = wsb[OFF_SUMS + s * 4 + 3] * inv;
    }
    wsb[OFF_MEAN + s * 4 + 0] = m0;
    wsb[OFF_MEAN + s * 4 + 1] = m1;
    wsb[OFF_MEAN + s * 4 + 2] = m2;
    wsb[OFF_MEAN + s * 4 + 3] = m3;
    if (s >= 1) {
      int i = s - 1;  // row of M
      s_m[i][0] = m0; s_m[i][1] = m1; s_m[i][2] = m2; s_m[i][3] = m3;
      s_n2[i] = m0 * m0 + m1 * m1 + m2 * m2 + m3 * m3;
    }
  }
  __syncthreads();

  // Gram = M(32x4) * M^T(4x32) as four 16x16 WMMA tiles, K=4, full f32.
  // A 16x4 layout: lane L holds row M=L%16; VGPR0 = K (0 | 2), VGPR1 = K (1 | 3)
  // B 4x16 layout: lane L holds col N=L%16; VGPR0 = K (0 | 2), VGPR1 = K (1 | 3)
  // C/D 16x16:     lane L col N=L%16; VGPR v = row (v | v+8)
  const int col  = lane & 15;
  const int half = lane >> 4;
  const int k0   = half ? 2 : 0;
  const int k1   = half ? 3 : 1;

  float acc_dis = 0.0f;
  for (int rt = 0; rt < 2; ++rt) {
    const int Mr = rt * 16 + col;
    v2f a;
    a.x = s_m[Mr][k0];
    a.y = s_m[Mr][k1];
    for (int ct = 0; ct < 2; ++ct) {
      const int Nc = ct * 16 + col;
      v2f bm;
      bm.x = s_m[Nc][k0];   // B[k][n] = M[n][k]
      bm.y = s_m[Nc][k1];
      v8f c = {};
      v8f g = __builtin_amdgcn_wmma_f32_16x16x4_f32(
          /*neg_a=*/false, a, /*neg_b=*/false, bm,
          /*c_mod=*/(short)0, c, /*reuse_a=*/false, /*reuse_b=*/false);
#pragma unroll
      for (int v = 0; v < 8; ++v) {
        const int i = rt * 16 + v + (half ? 8 : 0);
        const int j = Nc;
        if (i < K && j < K && i != j) {
          float pd2 = s_n2[i] + s_n2[j] - 2.0f * g[v];
          float pd  = sqrtf(fmaxf(pd2, 0.0f));
          float t   = fmaxf(2.0f * DELTA_D - pd, 0.0f);
          acc_dis  += logf(t * t + 1.0f);
        }
      }
    }
  }

  float reg = (lane < K) ? logf(sqrtf(s_n2[lane]) + 1.0f) : 0.0f;

  atomicAdd(&s_red[0], acc_dis);
  atomicAdd(&s_red[1], reg);
  __syncthreads();

  if (lane == 0) {
    float l_dis = (K > 1) ? s_red[0] / (float)(K * (K - 1)) : 0.0f;
    float l_reg = s_red[1] / (float)NL * 0.001f;
    wsb[OFF_LDR] = l_dis + l_reg;
  }
}

// Pass 2: per-pixel distance-to-mean value, segment-summed (labels >= 1 only;
// label 0 never contributes to l_agg since agg[1:] is taken).
__global__ void k_val(const float* __restrict__ emb,
                      const int* __restrict__ inst,
                      const float* __restrict__ tmask,
                      const int* __restrict__ nlp,
                      float* __restrict__ ws) {
  const int b = blockIdx.y;
  int NL = *nlp; NL = NL < 1 ? 1 : (NL > MAXL ? MAXL : NL);

  float* wsb = ws + (size_t)b * WS_PER_B;

  __shared__ float s_val[MAXL];
  __shared__ float s_mean[MAXL * 4];
  for (int i = threadIdx.x; i < MAXL; i += blockDim.x) s_val[i] = 0.0f;
  for (int i = threadIdx.x; i < MAXL * 4; i += blockDim.x)
    s_mean[i] = (i < NL * 4) ? wsb[OFF_MEAN + i] : 0.0f;
  __syncthreads();

  const float* e  = emb  + (size_t)b * DFEAT * HWPIX;
  const int*   ib = inst + (size_t)b * HWPIX;
  const float* tb = tmask + (size_t)b * HWPIX;

  const int stride = gridDim.x * blockDim.x;
  for (int n = blockIdx.x * blockDim.x + threadIdx.x; n < HWPIX; n += stride) {
    int lab = ib[n];
    int i2  = (tb[n] > 0.5f) ? lab : 0;
    if (i2 <= 0 || i2 >= NL) continue;
    float d0 = e[0 * HWPIX + n] - s_mean[i2 * 4 + 0];
    float d1 = e[1 * HWPIX + n] - s_mean[i2 * 4 + 1];
    float d2 = e[2 * HWPIX + n] - s_mean[i2 * 4 + 2];
    float d3 = e[3 * HWPIX + n] - s_mean[i2 * 4 + 3];
    float dist = sqrtf(d0 * d0 + d1 * d1 + d2 * d2 + d3 * d3);
    float t = fmaxf(dist - DELTA_V, 0.0f);
    atomicAdd(&s_val[i2], logf(t * t + 1.0f));
  }
  __syncthreads();

  for (int i = threadIdx.x; i < NL; i += blockDim.x)
    atomicAdd(&wsb[OFF_VALS + i], s_val[i]);
}

// Final: l_agg per batch + stored (l_dis + l_reg), mean over batch -> scalar.
__global__ void k_final(const int* __restrict__ nlp,
                        const float* __restrict__ ws,
                        float* __restrict__ out, int B) {
  int NL = *nlp; NL = NL < 1 ? 1 : (NL > MAXL ? MAXL : NL);
  const int K = NL - 1;

  __shared__ float s_acc;
  if (threadIdx.x == 0) s_acc = 0.0f;
  __syncthreads();

  float local = 0.0f;
  if (K > 0) {
    for (int idx = threadIdx.x; idx < B * K; idx += blockDim.x) {
      int b = idx / K;
      int s = idx % K + 1;
      const float* wsb = ws + (size_t)b * WS_PER_B;
      local += wsb[OFF_VALS + s] / fmaxf(wsb[OFF_CNTI + s], 1.0f) / (float)K;
    }
  }
  for (int b = threadIdx.x; b < B; b += blockDim.x)
    local += ws[(size_t)b * WS_PER_B + OFF_LDR];

  atomicAdd(&s_acc, local);
  __syncthreads();
  if (threadIdx.x == 0) out[0] = s_acc / (float)B;
}

extern "C" void kernel_launch(void* const* d_in, const int* in_sizes, int n_in,
                              void* d_out, int out_size, void* d_ws, size_t ws_size,
                              hipStream_t stream) {
  const float* emb   = (const float*)d_in[0];  // (B, 4, 640, 640)
  const int*   inst  = (const int*)d_in[1];    // (B, 640, 640)
  const float* kern  = (const float*)d_in[2];  // (B, 640, 640)
  const float* tmask = (const float*)d_in[3];  // (B, 640, 640)
  const int*   nlp   = (const int*)d_in[4];    // scalar num_labels

  const int B = in_sizes[1] / HWPIX;           // 8 for the reference shapes
  float* ws  = (float*)d_ws;
  float* out = (float*)d_out;

  // zero accumulators
  k_zero<<<dim3(8), dim3(256), 0, stream>>>(ws, B * WS_PER_B);

  // pass 1: segment sums + counts
  k_segsum<<<dim3(200, B), dim3(256), 0, stream>>>(emb, inst, kern, tmask, nlp, ws);

  // means + WMMA pairwise-distance loss (one wave per batch)
  k_mean_pdist<<<dim3(B), dim3(32), 0, stream>>>(nlp, ws);

  // pass 2: per-pixel val, segment-summed
  k_val<<<dim3(200, B), dim3(256), 0, stream>>>(emb, inst, tmask, nlp, ws);

  // final reduction -> scalar loss
  k_final<<<dim3(1), dim3(256), 0, stream>>>(nlp, ws, out, B);
}